// SimCLRLoss_35613868819238
// MI455X (gfx1250) — compile-verified
//
#include <hip/hip_runtime.h>
#include <math.h>

typedef float v2f __attribute__((ext_vector_type(2)));
typedef float v8f __attribute__((ext_vector_type(8)));

#define B_ROWS 4096
#define D_DIM  256
#define N_TOT  (2 * B_ROWS)            // 8192 rows of z
#define TINV   2.0f                    // 1 / temperature
#define E2     7.3890560989306495f     // exp(2) == diagonal term
#define EPS_DEN 1e-8f
#define EPS_NRM 1e-12f

#define BM 128
#define BN 128
#define BK 32
#define LDK 34                         // padded LDS row stride (even -> 8B aligned pairs)

// ---------------------------------------------------------------------------
// 1) Row L2-normalize both feature blocks into z[8192][256]
//    one wave32 per row, 8 floats (2 x float4) per lane
// ---------------------------------------------------------------------------
__global__ __launch_bounds__(256) void simclr_normalize(
    const float* __restrict__ f1, const float* __restrict__ f2,
    float* __restrict__ z) {
  const int row  = blockIdx.x * 8 + (threadIdx.x >> 5);
  const int lane = threadIdx.x & 31;
  const float* src = (row < B_ROWS) ? (f1 + (size_t)row * D_DIM)
                                    : (f2 + (size_t)(row - B_ROWS) * D_DIM);
  const float4 a = ((const float4*)src)[lane];
  const float4 b = ((const float4*)src)[lane + 32];
  float ss = a.x*a.x + a.y*a.y + a.z*a.z + a.w*a.w
           + b.x*b.x + b.y*b.y + b.z*b.z + b.w*b.w;
  #pragma unroll
  for (int m = 16; m >= 1; m >>= 1) ss += __shfl_xor(ss, m, 32);
  const float inv = 1.0f / fmaxf(sqrtf(ss), EPS_NRM);
  float4* dst = (float4*)(z + (size_t)row * D_DIM);
  dst[lane]      = make_float4(a.x*inv, a.y*inv, a.z*inv, a.w*inv);
  dst[lane + 32] = make_float4(b.x*inv, b.y*inv, b.z*inv, b.w*inv);
}

// ---------------------------------------------------------------------------
// 2) Zero the rowsum accumulator (needed every call: atomics accumulate)
// ---------------------------------------------------------------------------
__global__ void simclr_zero(float* __restrict__ p, int n) {
  int i = blockIdx.x * blockDim.x + threadIdx.x;
  if (i < n) p[i] = 0.0f;
}

// ---------------------------------------------------------------------------
// 3) Symmetric tiled f32 WMMA GEMM over the UPPER-TRIANGULAR tiles only:
//    C_tile = Z[bm..]*Z[bn..]^T for bn >= bm. exp(2*c) in-place, then
//    row sums -> rowsum[bm..], and (off-diag) column sums -> rowsum[bn..]
//    via the symmetry S[j][i] = S[i][j]. Global K-tile loads are register-
//    prefetched so they overlap the WMMA compute phase.
//    Block = 256 threads = 8 waves (4 M x 2 N); each wave owns 32x64 via
//    2x4 accumulators of v_wmma_f32_16x16x4_f32.
// ---------------------------------------------------------------------------
__global__ __launch_bounds__(256) void simclr_gemm_rowsum(
    const float* __restrict__ z, float* __restrict__ rowsum) {
  const int bmIdx = blockIdx.y;
  const int bnIdx = blockIdx.x;
  if (bnIdx < bmIdx) return;           // symmetry: upper-triangular tiles only

  __shared__ float As[BM * LDK];
  __shared__ float Bs[BN * LDK];

  const int tid  = threadIdx.x;
  const int lane = tid & 31;
  const int wave = tid >> 5;
  const int wm   = wave & 3;          // wave M position (0..3) -> 32 rows each
  const int wn   = wave >> 2;         // wave N position (0..1) -> 64 cols each
  const int bm   = bmIdx * BM;
  const int bn   = bnIdx * BN;

  v8f acc[2][4];
  #pragma unroll
  for (int mi = 0; mi < 2; ++mi)
    #pragma unroll
    for (int ni = 0; ni < 4; ++ni)
      acc[mi][ni] = (v8f)0.0f;

  // Register prefetch of the first K-tile (1024 float4 per matrix / 256 thr)
  float4 va[4], vb[4];
  #pragma unroll
  for (int i = 0; i < 4; ++i) {
    const int q  = tid + i * 256;
    const int r  = q >> 3;
    const int c4 = q & 7;
    va[i] = ((const float4*)(z + (size_t)(bm + r) * D_DIM))[c4];
    vb[i] = ((const float4*)(z + (size_t)(bn + r) * D_DIM))[c4];
  }

  for (int kt = 0; kt < D_DIM; kt += BK) {
    // Commit prefetched registers to LDS
    #pragma unroll
    for (int i = 0; i < 4; ++i) {
      const int q  = tid + i * 256;
      const int r  = q >> 3;
      const int c4 = q & 7;
      float* ap = &As[r * LDK + c4 * 4];
      ap[0] = va[i].x; ap[1] = va[i].y; ap[2] = va[i].z; ap[3] = va[i].w;
      float* bp = &Bs[r * LDK + c4 * 4];
      bp[0] = vb[i].x; bp[1] = vb[i].y; bp[2] = vb[i].z; bp[3] = vb[i].w;
    }
    __syncthreads();

    // Prefetch NEXT K-tile while the WMMA phase below runs
    if (kt + BK < D_DIM) {
      #pragma unroll
      for (int i = 0; i < 4; ++i) {
        const int q  = tid + i * 256;
        const int r  = q >> 3;
        const int c4 = q & 7;
        va[i] = ((const float4*)(z + (size_t)(bm + r) * D_DIM + kt + BK))[c4];
        vb[i] = ((const float4*)(z + (size_t)(bn + r) * D_DIM + kt + BK))[c4];
      }
    }

    const int half = lane >> 4;       // 0: K pair {k,k+1}, 1: {k+2,k+3}
    const int r16  = lane & 15;
    #pragma unroll
    for (int k = 0; k < BK; k += 4) {
      const int kk = k + 2 * half;
      v2f afrag[2], bfrag[4];
      #pragma unroll
      for (int mi = 0; mi < 2; ++mi) {
        const int mrow = wm * 32 + mi * 16 + r16;
        afrag[mi].x = As[mrow * LDK + kk];
        afrag[mi].y = As[mrow * LDK + kk + 1];
      }
      #pragma unroll
      for (int ni = 0; ni < 4; ++ni) {
        const int nrow = wn * 64 + ni * 16 + r16;
        bfrag[ni].x = Bs[nrow * LDK + kk];
        bfrag[ni].y = Bs[nrow * LDK + kk + 1];
      }
      #pragma unroll
      for (int mi = 0; mi < 2; ++mi)
        #pragma unroll
        for (int ni = 0; ni < 4; ++ni)
          acc[mi][ni] = __builtin_amdgcn_wmma_f32_16x16x4_f32(
              false, afrag[mi], false, bfrag[ni],
              (short)0, acc[mi][ni], false, false);
    }
    __syncthreads();
  }

  // ---- Epilogue -----------------------------------------------------------
  // exp(2*c) in place: each element's exp is shared by row & column sums.
  #pragma unroll
  for (int mi = 0; mi < 2; ++mi)
    #pragma unroll
    for (int ni = 0; ni < 4; ++ni)
      #pragma unroll
      for (int r = 0; r < 8; ++r)
        acc[mi][ni][r] = __expf(TINV * acc[mi][ni][r]);

  // Row sums over this block's 128 columns.
  // C/D layout: VGPR r holds row r (lanes 0-15) / row r+8 (lanes 16-31),
  // column = lane % 16 -> xor-reduce within each 16-lane half.
  #pragma unroll
  for (int mi = 0; mi < 2; ++mi) {
    #pragma unroll
    for (int r = 0; r < 8; ++r) {
      float v = acc[mi][0][r] + acc[mi][1][r] + acc[mi][2][r] + acc[mi][3][r];
      #pragma unroll
      for (int m = 8; m >= 1; m >>= 1) v += __shfl_xor(v, m, 32);
      if ((lane & 15) == 0) {
        const int grow = bm + wm * 32 + mi * 16 + (lane >> 4) * 8 + r;
        atomicAdd(&rowsum[grow], v);
      }
    }
  }

  // Column sums (off-diagonal only): S[j][i] contributions for rows bn..
  // Column n lives in lane n (rows 0-7) and lane n+16 (rows 8-15).
  if (bm != bn) {
    #pragma unroll
    for (int ni = 0; ni < 4; ++ni) {
      float s = 0.0f;
      #pragma unroll
      for (int mi = 0; mi < 2; ++mi)
        #pragma unroll
        for (int r = 0; r < 8; ++r)
          s += acc[mi][ni][r];
      s += __shfl_xor(s, 16, 32);     // combine the two 8-row halves
      if (lane < 16)
        atomicAdd(&rowsum[bn + wn * 64 + ni * 16 + lane], s);
    }
  }
}

// ---------------------------------------------------------------------------
// 4) Per-row loss: numerator = exp(2 * <z_i, z_{(i+B)%2B}>)  =>
//    loss_i = log(max(rowsum_i - e^2, eps)) - 2*cos_i ; block partial sums.
// ---------------------------------------------------------------------------
__global__ __launch_bounds__(256) void simclr_loss(
    const float* __restrict__ z, const float* __restrict__ rowsum,
    float* __restrict__ partials) {
  __shared__ float sdata[8];
  const int row  = blockIdx.x * 8 + (threadIdx.x >> 5);
  const int lane = threadIdx.x & 31;
  const int pos  = (row < B_ROWS) ? (row + B_ROWS) : (row - B_ROWS);
  const float4* zi = (const float4*)(z + (size_t)row * D_DIM);
  const float4* zj = (const float4*)(z + (size_t)pos * D_DIM);
  const float4 a0 = zi[lane], a1 = zi[lane + 32];
  const float4 b0 = zj[lane], b1 = zj[lane + 32];
  float dot = a0.x*b0.x + a0.y*b0.y + a0.z*b0.z + a0.w*b0.w
            + a1.x*b1.x + a1.y*b1.y + a1.z*b1.z + a1.w*b1.w;
  #pragma unroll
  for (int m = 16; m >= 1; m >>= 1) dot += __shfl_xor(dot, m, 32);
  if (lane == 0) {
    const float den = fmaxf(rowsum[row] - E2, EPS_DEN);
    sdata[threadIdx.x >> 5] = logf(den) - TINV * dot;
  }
  __syncthreads();
  if (threadIdx.x == 0) {
    float s = 0.0f;
    #pragma unroll
    for (int i = 0; i < 8; ++i) s += sdata[i];
    partials[blockIdx.x] = s;
  }
}

// ---------------------------------------------------------------------------
// 5) Final reduction of 1024 block partials -> scalar mean
// ---------------------------------------------------------------------------
__global__ __launch_bounds__(1024) void simclr_final(
    const float* __restrict__ partials, float* __restrict__ out) {
  __shared__ float sdata[1024];
  const int t = threadIdx.x;
  sdata[t] = partials[t];
  __syncthreads();
  for (int s = 512; s > 0; s >>= 1) {
    if (t < s) sdata[t] += sdata[t + s];
    __syncthreads();
  }
  if (t == 0) out[0] = sdata[0] / (float)N_TOT;
}

// ---------------------------------------------------------------------------
extern "C" void kernel_launch(void* const* d_in, const int* in_sizes, int n_in,
                              void* d_out, int out_size, void* d_ws, size_t ws_size,
                              hipStream_t stream) {
  (void)in_sizes; (void)n_in; (void)out_size; (void)ws_size;
  const float* f1 = (const float*)d_in[0];
  const float* f2 = (const float*)d_in[1];
  float* out = (float*)d_out;

  char*  ws       = (char*)d_ws;
  float* z        = (float*)ws;                                     // 8 MB
  float* rowsum   = (float*)(ws + (size_t)N_TOT * D_DIM * sizeof(float));
  float* partials = rowsum + N_TOT;                                 // 1024 floats

  simclr_normalize<<<N_TOT / 8, 256, 0, stream>>>(f1, f2, z);
  simclr_zero<<<(N_TOT + 255) / 256, 256, 0, stream>>>(rowsum, N_TOT);
  dim3 grid(N_TOT / BN, N_TOT / BM);                                // 64 x 64
  simclr_gemm_rowsum<<<grid, 256, 0, stream>>>(z, rowsum);
  simclr_loss<<<N_TOT / 8, 256, 0, stream>>>(z, rowsum, partials);
  simclr_final<<<1, 1024, 0, stream>>>(partials, out);
}